// VectorQuantizerWithWeakTrick_11166914969841
// MI455X (gfx1250) — compile-verified
//
#include <hip/hip_runtime.h>
#include <hip/hip_bf16.h>
#include <math.h>

typedef __attribute__((ext_vector_type(2))) float v2f;
typedef __attribute__((ext_vector_type(4))) float v4f;
typedef __attribute__((ext_vector_type(8))) float v8f;
typedef int v4i_vs __attribute__((vector_size(16)));   // matches builtin's V4i
typedef __attribute__((address_space(1))) v4i_vs* gptr_v4i;
typedef __attribute__((address_space(3))) v4i_vs* lptr_v4i;

#define N_E    1024
#define E_DIM  64
#define HW     4096            // 64*64
#define N_TOK  65536           // 16*4096 tokens
#define ZQ_ELEMS (N_TOK * E_DIM)

#define CHUNK_CODES 64                    // codes staged per LDS chunk
#define ROW_F       68                    // padded LDS row stride (floats) -> no ds bank conflicts
#define N_CHUNKS    (N_E / CHUNK_CODES)   // 16

#if defined(__gfx1250__) && __has_builtin(__builtin_amdgcn_global_load_async_to_lds_b128)
#define USE_ASYNC_LDS 1
#else
#define USE_ASYNC_LDS 0
#endif

__device__ __forceinline__ void wait_async_zero() {
#if __has_builtin(__builtin_amdgcn_s_wait_asynccnt)
  __builtin_amdgcn_s_wait_asynccnt(0);
#else
  asm volatile("s_wait_asynccnt 0" ::: "memory");
#endif
}

// ---------------- workspace layout (bytes) ----------------
// [0,    4096)  float e2[1024]      code squared norms
// [4096, 8192)  int   counts[1024]  usage histogram
// [8192, 8196)  float sse           sum of squared errors

__global__ void vq_prep(const float* __restrict__ emb,
                        float* __restrict__ e2,
                        int* __restrict__ counts,
                        float* __restrict__ sse) {
  int tid = blockIdx.x * blockDim.x + threadIdx.x;
  if (tid < N_E) {
    const float* row = emb + tid * E_DIM;
    float s = 0.f;
    #pragma unroll
    for (int c = 0; c < E_DIM; ++c) s += row[c] * row[c];
    e2[tid] = s;
    counts[tid] = 0;
    if (tid == 0) *sse = 0.f;
  }
}

// One wave handles 16 tokens; block = 8 waves = 128 tokens; grid = 512 blocks.
__launch_bounds__(256)
__global__ void vq_main(const float* __restrict__ z,
                        const float* __restrict__ emb,
                        const float* __restrict__ e2,
                        int* __restrict__ counts,
                        float* __restrict__ sse,
                        float* __restrict__ zq_out,
                        int* __restrict__ idx_out) {
  // double-buffered codebook chunk: 2 x 64 codes x 68 floats (padded) = 34 KB
  __shared__ float sEmb[2 * CHUNK_CODES * ROW_F];
  __shared__ int   sIdx[8 * 16];           // winning code per token, per wave

  const int tid   = threadIdx.x;
  const int lane  = tid & 31;
  const int wave  = tid >> 5;
  const int gwave = blockIdx.x * 8 + wave;
  const int tok0  = gwave * 16;          // first token of this wave's tile
  const int b     = tok0 >> 12;          // /4096 (tokens per image)
  const int hw    = tok0 & 4095;         // contiguous 16-token span (4096%16==0)
  const int m     = lane & 15;           // row (token) for A / col (code) for B,C
  const int h     = lane >> 4;           // half-wave selects K pair
  const int zbase = b * (E_DIM * HW) + hw;

  // cooperative-copy role: each thread moves 4x float4 per chunk
  const int crow = tid >> 4;             // 0..15 (base row of sweep)
  const int ccol = (tid & 15) * 4;       // float column within a code row

  // ---- A panel: 16 tokens x 64 channels, resident in 32 VGPRs ----
  // ISA 16x4 fp32 A layout: lanes 0-15 hold K={0,1}, lanes 16-31 K={2,3}.
  v2f a[16];
  #pragma unroll
  for (int s = 0; s < 16; ++s) {
    int c0 = 4 * s + 2 * h;
    a[s].x = z[zbase + (c0 + 0) * HW + m];
    a[s].y = z[zbase + (c0 + 1) * HW + m];
  }

  float minv[8];
  int   mini[8];
  #pragma unroll
  for (int r = 0; r < 8; ++r) { minv[r] = __builtin_inff(); mini[r] = 0; }

#if USE_ASYNC_LDS
  // ---- async global -> LDS staging (ASYNCcnt path) ----
  // builtin signature: (global int4* src, shared int4* dst, imm offset, imm cpol)
  #define ISSUE_CHUNK(chunk, buf)                                              \
    {                                                                          \
      _Pragma("unroll")                                                        \
      for (int j = 0; j < 4; ++j) {                                            \
        int row = j * 16 + crow;                                               \
        const float* g = emb + ((chunk) * CHUNK_CODES + row) * E_DIM + ccol;   \
        float* l = &sEmb[(buf) * CHUNK_CODES * ROW_F + row * ROW_F + ccol];    \
        __builtin_amdgcn_global_load_async_to_lds_b128(                        \
            (gptr_v4i)g, (lptr_v4i)l, 0, 0);                                   \
      }                                                                        \
    }
  ISSUE_CHUNK(0, 0)
#else
  // ---- fallback staging: global b128 -> regs -> ds_store_b128 ----
  v4f stageReg[4];
  #pragma unroll
  for (int j = 0; j < 4; ++j)
    stageReg[j] = *(const v4f*)(emb + (j * 16 + crow) * E_DIM + ccol);
#endif

  for (int chunk = 0; chunk < N_CHUNKS; ++chunk) {
    const int buf = chunk & 1;
#if USE_ASYNC_LDS
    wait_async_zero();               // my async copies for this chunk landed
    __syncthreads();                 // => everyone's landed; prev buffer free
    if (chunk + 1 < N_CHUNKS) ISSUE_CHUNK(chunk + 1, buf ^ 1)
#else
    #pragma unroll
    for (int j = 0; j < 4; ++j)
      *(v4f*)&sEmb[buf * CHUNK_CODES * ROW_F + (j * 16 + crow) * ROW_F + ccol] =
          stageReg[j];
    __syncthreads();
    if (chunk + 1 < N_CHUNKS) {
      #pragma unroll
      for (int j = 0; j < 4; ++j)
        stageReg[j] = *(const v4f*)(emb +
            ((chunk + 1) * CHUNK_CODES + j * 16 + crow) * E_DIM + ccol);
    }
#endif

    const float* bchunk = &sEmb[buf * CHUNK_CODES * ROW_F];
    #pragma unroll
    for (int sub = 0; sub < 4; ++sub) {
      const int nc = sub * 16 + m;                 // code within chunk (lane col)
      const int n  = chunk * CHUNK_CODES + nc;     // global code id
      const float e2n = e2[n];
      const float* brow = bchunk + nc * ROW_F + 2 * h;
      v8f c = {0.f, 0.f, 0.f, 0.f, 0.f, 0.f, 0.f, 0.f};
      #pragma unroll
      for (int s = 0; s < 16; ++s) {
        v2f bf = *(const v2f*)(brow + 4 * s);      // ds_load_b64, conflict-free
        c = __builtin_amdgcn_wmma_f32_16x16x4_f32(false, a[s], false, bf,
                                                  (short)0, c, false, false);
      }
      #pragma unroll
      for (int r = 0; r < 8; ++r) {
        float d = e2n - 2.0f * c[r];
        if (d < minv[r]) { minv[r] = d; mini[r] = n; } // strict < keeps lowest n
      }
    }
  }

  // ---- argmin across the 16 columns (lanes within each half-wave) ----
  #pragma unroll
  for (int off = 1; off < 16; off <<= 1) {
    #pragma unroll
    for (int r = 0; r < 8; ++r) {
      float ov = __shfl_xor(minv[r], off, 32);
      int   oi = __shfl_xor(mini[r], off, 32);
      if (ov < minv[r] || (ov == minv[r] && oi < mini[r])) {
        minv[r] = ov; mini[r] = oi;
      }
    }
  }

  // lanes 0 (rows 0-7) and 16 (rows 8-15) hold the winners
  if (m == 0) {
    #pragma unroll
    for (int r = 0; r < 8; ++r) {
      int row  = r + 8 * h;
      int code = mini[r];
      sIdx[wave * 16 + row] = code;
      idx_out[tok0 + row]   = code;
      atomicAdd(&counts[code], 1);
    }
  }
  __syncthreads();

  // ---- gather z_q (NCHW), accumulate SSE for the loss ----
  float acc = 0.f;
  #pragma unroll
  for (int i = 0; i < 32; ++i) {
    int f  = i * 32 + lane;        // f = c*16 + mt  (c-major so writes coalesce)
    int mt = f & 15;
    int c  = f >> 4;
    int code = sIdx[wave * 16 + mt];
    int addr = zbase + c * HW + mt;
    float q  = emb[code * E_DIM + c];
    float d  = q - z[addr];
    zq_out[addr] = q;              // straight-through: forward value is z_q
    acc += d * d;
  }
  #pragma unroll
  for (int off = 1; off < 32; off <<= 1) acc += __shfl_xor(acc, off, 32);
  if (lane == 0) atomicAdd(sse, acc);
}

__global__ void vq_final(const int* __restrict__ counts,
                         const float* __restrict__ sse,
                         float* __restrict__ loss_out,
                         float* __restrict__ perp_out) {
  __shared__ float red[32];
  int tid = threadIdx.x;                               // 1024 threads
  float p = (float)counts[tid] * (1.0f / (float)N_TOK);
  float v = p * logf(p + 1e-10f);
  #pragma unroll
  for (int off = 1; off < 32; off <<= 1) v += __shfl_xor(v, off, 32);
  if ((tid & 31) == 0) red[tid >> 5] = v;
  __syncthreads();
  if (tid < 32) {
    float s = red[tid];
    #pragma unroll
    for (int off = 1; off < 32; off <<= 1) s += __shfl_xor(s, off, 32);
    if (tid == 0) {
      *perp_out = expf(-s);
      // loss = (1 + beta) * mean(diff^2), beta = 0.25, over 16*64*64*64 elems
      *loss_out = 1.25f * (*sse) / 4194304.0f;
    }
  }
}

extern "C" void kernel_launch(void* const* d_in, const int* in_sizes, int n_in,
                              void* d_out, int out_size, void* d_ws, size_t ws_size,
                              hipStream_t stream) {
  const float* z   = (const float*)d_in[0];   // [16,64,64,64] NCHW fp32
  const float* emb = (const float*)d_in[1];   // [1024,64] fp32

  float* e2     = (float*)d_ws;
  int*   counts = (int*)((char*)d_ws + 4096);
  float* sse    = (float*)((char*)d_ws + 8192);

  float* out  = (float*)d_out;
  float* zq   = out;                               // [16,64,64,64]
  float* loss = out + ZQ_ELEMS;                    // scalar
  float* perp = out + ZQ_ELEMS + 1;                // scalar
  int*   idxo = (int*)(out + ZQ_ELEMS + 2);        // [16,4096] int32

  vq_prep <<<4,   256, 0, stream>>>(emb, e2, counts, sse);
  vq_main <<<512, 256, 0, stream>>>(z, emb, e2, counts, sse, zq, idxo);
  vq_final<<<1,  1024, 0, stream>>>(counts, sse, loss, perp);
}